// NodeEmbedder_27479200760104
// MI455X (gfx1250) — compile-verified
//
#include <hip/hip_runtime.h>
#include <hip/hip_bf16.h>

// ---------------------------------------------------------------------------
// Model constants (from the reference)
// ---------------------------------------------------------------------------
#define BB 4
#define TT 128
#define NN 256
#define HH 64
#define DI 128
#define DS 16
#define DC 3
#define DTR 4
#define NSEQ (BB * NN)   // 1024 sequences

typedef float v2f __attribute__((ext_vector_type(2)));
typedef float v8f __attribute__((ext_vector_type(8)));

// D = A(16x4,f32) * B(4x16,f32) + C(16x16,f32)  -- exact fp32 matrix op
static __device__ __forceinline__ v8f wmma_f32(v2f a, v2f b, v8f c) {
    return __builtin_amdgcn_wmma_f32_16x16x4_f32(
        /*neg_a=*/false, a, /*neg_b=*/false, b,
        /*c_mod=*/(short)0, c, /*reuse_a=*/false, /*reuse_b=*/false);
}

static __device__ __forceinline__ float silu_f(float v) {
    return v / (1.0f + __expf(-v));
}

// ---------------------------------------------------------------------------
// K1: fused graph encoder.
//   emb = relu(G @ W1^T + b1) @ W2^T + b2, written transposed into seq layout:
//   seq[(b*N + node)*T*H + t*H + h]
// grid = B*T blocks, 256 threads (8 waves). Each wave computes two 16x64
// strips: stage-1 WMMA (K=256), relu+bias into wave-private LDS, stage-2
// WMMA (K=64) reading A-fragments from LDS.
// ---------------------------------------------------------------------------
__global__ __launch_bounds__(256) void k_encoder(
    const float* __restrict__ G, const float* __restrict__ W1,
    const float* __restrict__ b1, const float* __restrict__ W2,
    const float* __restrict__ b2, float* __restrict__ seq)
{
    __shared__ float lds_h1[8][16 * HH];   // 8 waves * 4KB = 32KB

    const int bt   = blockIdx.x;
    const int b    = bt >> 7;              // bt / T
    const int t    = bt & (TT - 1);
    const int wave = threadIdx.x >> 5;
    const int lane = threadIdx.x & 31;
    const int lh   = lane >> 4;            // 0: K lo pair, 1: K hi pair
    const int l16  = lane & 15;

    const float* Gbt = G + (size_t)bt * NN * NN;
    float* lds = &lds_h1[wave][0];

    for (int strip = wave; strip < 16; strip += 8) {
        const int m0 = strip * 16;

        // ---- stage 1: h1 strip (16 x 64) = G[m0:m0+16,:] @ W1^T ----
        v8f acc[4] = {};
        for (int k0 = 0; k0 < NN; k0 += 4) {
            const float* ga = Gbt + (size_t)(m0 + l16) * NN + k0 + lh * 2;
            v2f a; a.x = ga[0]; a.y = ga[1];
            #pragma unroll
            for (int nt = 0; nt < 4; ++nt) {
                const float* wb = W1 + (size_t)(nt * 16 + l16) * NN + k0 + lh * 2;
                v2f bf; bf.x = wb[0]; bf.y = wb[1];
                acc[nt] = wmma_f32(a, bf, acc[nt]);
            }
        }
        // relu + bias -> LDS (row-major 16 x 64, wave private -> no barrier:
        // LDS ops are in-order within a wave)
        #pragma unroll
        for (int nt = 0; nt < 4; ++nt) {
            const float bias = b1[nt * 16 + l16];
            #pragma unroll
            for (int v = 0; v < 8; ++v) {
                float val = acc[nt][v] + bias;
                val = val > 0.0f ? val : 0.0f;
                const int m = v + 8 * lh;
                lds[m * HH + nt * 16 + l16] = val;
            }
        }

        // ---- stage 2: emb strip (16 x 64) = h1 @ W2^T ----
        v8f e[4] = {};
        for (int k0 = 0; k0 < HH; k0 += 4) {
            const float* la = lds + l16 * HH + k0 + lh * 2;
            v2f a; a.x = la[0]; a.y = la[1];
            #pragma unroll
            for (int gt = 0; gt < 4; ++gt) {
                const float* wb = W2 + (size_t)(gt * 16 + l16) * HH + k0 + lh * 2;
                v2f bf; bf.x = wb[0]; bf.y = wb[1];
                e[gt] = wmma_f32(a, bf, e[gt]);
            }
        }
        // + b2, scatter into sequence-major layout
        #pragma unroll
        for (int gt = 0; gt < 4; ++gt) {
            const float bias = b2[gt * 16 + l16];
            #pragma unroll
            for (int v = 0; v < 8; ++v) {
                const int node = m0 + v + 8 * lh;
                const int h    = gt * 16 + l16;
                seq[((size_t)(b * NN + node) * TT + t) * HH + h] = e[gt][v] + bias;
            }
        }
    }
}

// ---------------------------------------------------------------------------
// K2: in_proj + causal conv(3) + SiLU, per sequence.
//   xz = seq_s(128x64) @ Wip^T(64x256); x part staged in 64KB dynamic LDS,
//   z kept only at t=T-1; then conv+silu -> xs (global).
// grid = 1024 blocks, 256 threads; wave w owns d-columns [w*32, w*32+32).
// ---------------------------------------------------------------------------
__global__ __launch_bounds__(256) void k_inproj_conv(
    const float* __restrict__ seq, const float* __restrict__ Wip,
    const float* __restrict__ cw, const float* __restrict__ cb,
    float* __restrict__ xs, float* __restrict__ z_last)
{
    extern __shared__ float xL[];          // TT * DI = 128*128 floats = 64KB

    const int s    = blockIdx.x;
    const int wave = threadIdx.x >> 5;
    const int lane = threadIdx.x & 31;
    const int lh   = lane >> 4;
    const int l16  = lane & 15;

    const float* S = seq + (size_t)s * TT * HH;

    for (int mt = 0; mt < 8; ++mt) {       // 8 tiles of t
        v8f acc[2] = {};
        for (int k0 = 0; k0 < HH; k0 += 4) {
            const float* sa = S + (size_t)(mt * 16 + l16) * HH + k0 + lh * 2;
            v2f a; a.x = sa[0]; a.y = sa[1];
            #pragma unroll
            for (int j = 0; j < 2; ++j) {
                const int dcol = (wave * 2 + j) * 16 + l16;
                const float* wb = Wip + (size_t)dcol * HH + k0 + lh * 2;
                v2f bf; bf.x = wb[0]; bf.y = wb[1];
                acc[j] = wmma_f32(a, bf, acc[j]);
            }
        }
        #pragma unroll
        for (int j = 0; j < 2; ++j) {
            const int dcol = (wave * 2 + j) * 16 + l16;
            #pragma unroll
            for (int v = 0; v < 8; ++v) {
                const int tt = mt * 16 + v + 8 * lh;
                const float val = acc[j][v];
                if (dcol < DI) {
                    xL[tt * DI + dcol] = val;           // x half -> LDS
                } else if (tt == TT - 1) {
                    z_last[(size_t)s * DI + (dcol - DI)] = val;  // z, last t only
                }
            }
        }
    }
    __syncthreads();

    // causal conv (DC=3) + SiLU from LDS -> global xs
    for (int e = 0; e < (TT * DI) / 256; ++e) {
        const int idx = e * 256 + threadIdx.x;
        const int t = idx >> 7;            // / DI
        const int d = idx & (DI - 1);
        float a = cb[d];
        #pragma unroll
        for (int k = 0; k < DC; ++k) {
            const int tt = t - (DC - 1) + k;
            const float xv = (tt >= 0) ? xL[tt * DI + d] : 0.0f;
            a += xv * cw[d * DC + k];
        }
        xs[(size_t)s * TT * DI + idx] = silu_f(a);
    }
}

// ---------------------------------------------------------------------------
// K3: x_proj. x_dbl(128 x 36, padded to 48) = xs_s(128x128) @ Xp^T(128x36).
// grid = 1024 blocks, 256 threads; wave = t-tile (8 tiles), 3 N-tiles each.
// ---------------------------------------------------------------------------
#define EPAD 48
__global__ __launch_bounds__(256) void k_xproj(
    const float* __restrict__ xs, const float* __restrict__ Xp,
    float* __restrict__ xd)
{
    const int s    = blockIdx.x;
    const int mt   = threadIdx.x >> 5;     // wave id = t tile
    const int lane = threadIdx.x & 31;
    const int lh   = lane >> 4;
    const int l16  = lane & 15;

    const float* X = xs + (size_t)s * TT * DI;

    v8f acc[3] = {};
    for (int k0 = 0; k0 < DI; k0 += 4) {
        const float* xa = X + (size_t)(mt * 16 + l16) * DI + k0 + lh * 2;
        v2f a; a.x = xa[0]; a.y = xa[1];
        #pragma unroll
        for (int nt = 0; nt < 3; ++nt) {
            const int e = nt * 16 + l16;
            v2f bf;
            if (e < DTR + 2 * DS) {        // 36 valid rows, pad rest with 0
                const float* wb = Xp + (size_t)e * DI + k0 + lh * 2;
                bf.x = wb[0]; bf.y = wb[1];
            } else {
                bf.x = 0.0f; bf.y = 0.0f;
            }
            acc[nt] = wmma_f32(a, bf, acc[nt]);
        }
    }
    #pragma unroll
    for (int nt = 0; nt < 3; ++nt) {
        #pragma unroll
        for (int v = 0; v < 8; ++v) {
            const int t = mt * 16 + v + 8 * lh;
            xd[((size_t)s * TT + t) * EPAD + nt * 16 + l16] = acc[nt][v];
        }
    }
}

// ---------------------------------------------------------------------------
// K4: selective scan. One block per sequence, one thread per channel d.
//   dt = softplus(x_dbl[:, :4] @ dtw[d]^T + dtb[d]) folded in.
//   h[16] kept in registers; y emitted only at t = T-1, times silu(z_last).
// ---------------------------------------------------------------------------
__global__ __launch_bounds__(DI) void k_scan(
    const float* __restrict__ xs, const float* __restrict__ xd,
    const float* __restrict__ dtw, const float* __restrict__ dtb,
    const float* __restrict__ A_log, const float* __restrict__ Dp,
    const float* __restrict__ z_last, float* __restrict__ ymod)
{
    const int s = blockIdx.x;
    const int d = threadIdx.x;

    float Areg[DS], h[DS];
    #pragma unroll
    for (int i = 0; i < DS; ++i) {
        Areg[i] = -__expf(A_log[d * DS + i]);
        h[i] = 0.0f;
    }
    const float w0 = dtw[d * DTR + 0], w1 = dtw[d * DTR + 1];
    const float w2 = dtw[d * DTR + 2], w3 = dtw[d * DTR + 3];
    const float bd = dtb[d];
    const float Dd = Dp[d];

    const float* XD = xd + (size_t)s * TT * EPAD;
    const float* XS = xs + (size_t)s * TT * DI;

    float ylast = 0.0f;
    for (int t = 0; t < TT; ++t) {
        const float* row = XD + t * EPAD;
        float dtr = row[0] * w0 + row[1] * w1 + row[2] * w2 + row[3] * w3 + bd;
        const float dt = (dtr > 20.0f) ? dtr : __logf(1.0f + __expf(dtr));
        const float xv  = XS[t * DI + d];
        const float dtx = dt * xv;
        #pragma unroll
        for (int i = 0; i < DS; ++i) {
            const float dA = __expf(dt * Areg[i]);
            h[i] = h[i] * dA + dtx * row[DTR + i];          // B_t[i]
        }
        if (t == TT - 1) {
            float y = Dd * xv;
            #pragma unroll
            for (int i = 0; i < DS; ++i) y += h[i] * row[DTR + DS + i];  // C_t[i]
            ylast = y;
        }
    }
    const float zv = z_last[(size_t)s * DI + d];
    ymod[(size_t)s * DI + d] = ylast * silu_f(zv);
}

// ---------------------------------------------------------------------------
// K5: out_proj. out[bn, h] = sum_d ymod[bn, d] * Wout[h, d]   (tiny: 16 MFLOP)
// ---------------------------------------------------------------------------
__global__ __launch_bounds__(HH) void k_outproj(
    const float* __restrict__ ymod, const float* __restrict__ Wout,
    float* __restrict__ out)
{
    const int bn = blockIdx.x;
    const int hh = threadIdx.x;
    const float* y = ymod + (size_t)bn * DI;
    const float* w = Wout + (size_t)hh * DI;
    float acc = 0.0f;
    #pragma unroll 8
    for (int d = 0; d < DI; ++d) acc += y[d] * w[d];
    out[(size_t)bn * HH + hh] = acc;
}

// ---------------------------------------------------------------------------
// Launch
// ---------------------------------------------------------------------------
extern "C" void kernel_launch(void* const* d_in, const int* in_sizes, int n_in,
                              void* d_out, int out_size, void* d_ws, size_t ws_size,
                              hipStream_t stream) {
    const float* G      = (const float*)d_in[0];   // (B,T,N,N)
    const float* W1     = (const float*)d_in[1];   // (H,N)
    const float* b1     = (const float*)d_in[2];   // (H,)
    const float* W2     = (const float*)d_in[3];   // (H,H)
    const float* b2     = (const float*)d_in[4];   // (H,)
    const float* Wip    = (const float*)d_in[5];   // (2*DI,H)
    const float* cw     = (const float*)d_in[6];   // (DI,DC)
    const float* cb     = (const float*)d_in[7];   // (DI,)
    const float* Xp     = (const float*)d_in[8];   // (DTR+2*DS, DI)
    const float* dtw    = (const float*)d_in[9];   // (DI,DTR)
    const float* dtb    = (const float*)d_in[10];  // (DI,)
    const float* A_log  = (const float*)d_in[11];  // (DI,DS)
    const float* Dp     = (const float*)d_in[12];  // (DI,)
    const float* Wout   = (const float*)d_in[13];  // (H,DI)
    float* out = (float*)d_out;

    float* ws = (float*)d_ws;
    // workspace layout (floats):
    //   [0 .. 8388608)         seq   (NSEQ*T*H)  -- dead after K2, reused:
    //   [0 .. 6291456)         xd    (NSEQ*T*48) -- written by K3
    //   [8388608 .. 25165824)  xs    (NSEQ*T*DI)
    //   [25165824 ..]          z_last (NSEQ*DI), ymod (NSEQ*DI)
    float* seq    = ws;
    float* xd     = ws;                                  // reuse of seq region
    float* xs     = ws + (size_t)NSEQ * TT * HH;
    float* z_last = xs + (size_t)NSEQ * TT * DI;
    float* ymod   = z_last + (size_t)NSEQ * DI;

    k_encoder<<<BB * TT, 256, 0, stream>>>(G, W1, b1, W2, b2, seq);
    k_inproj_conv<<<NSEQ, 256, TT * DI * sizeof(float), stream>>>(
        seq, Wip, cw, cb, xs, z_last);
    k_xproj<<<NSEQ, 256, 0, stream>>>(xs, Xp, xd);
    k_scan<<<NSEQ, DI, 0, stream>>>(xs, xd, dtw, dtb, A_log, Dp, z_last, ymod);
    k_outproj<<<NSEQ, HH, 0, stream>>>(ymod, Wout, out);
}